// ForwardDistance_10385230922577
// MI455X (gfx1250) — compile-verified
//
#include <hip/hip_runtime.h>

typedef __attribute__((ext_vector_type(2))) float v2f;
typedef __attribute__((ext_vector_type(8))) float v8f;

#define K_LDIM 512
#define K_AGG  64
#define ROWS   4096      // 4 * 1024
#define NPERB  64        // n-rows per block in phase 2

// ---------------- fast tanh: use gfx1250 hardware v_tanh_f32 if available ----
#if __has_builtin(__builtin_amdgcn_tanhf)
static __device__ __forceinline__ float fast_tanh(float x) {
  return __builtin_amdgcn_tanhf(x);
}
#elif __has_builtin(__builtin_amdgcn_tanh_f32)
static __device__ __forceinline__ float fast_tanh(float x) {
  return __builtin_amdgcn_tanh_f32(x);
}
#else
static __device__ __forceinline__ float fast_tanh(float x) { return tanhf(x); }
#endif

// ---------------------------------------------------------------------------
// Phase 2 (placed first so its asm shows in the snippet):
//   out[b,n,m] = sum_a tanh(DL[b,n,a] + CL[b,m,a]) * agg[a]
// Each thread owns one m (cl row in registers; agg is uniform -> SGPRs);
// each block stages NPERB dl rows into LDS (16 KB) and loops n over them.
// Writes are coalesced across m.
// ---------------------------------------------------------------------------
__global__ __launch_bounds__(256) void pairwise_tanh(
    const float* __restrict__ DL, const float* __restrict__ CL,
    const float* __restrict__ aggv, float* __restrict__ out) {
  __shared__ float4 sdl4[NPERB * (K_AGG / 4)];   // 64 rows x 64 f32 = 16 KB

  const int tid = threadIdx.x;
  const int b   = blockIdx.z;
  const int m   = blockIdx.y * 256 + tid;
  const int n0  = blockIdx.x * NPERB;

  // stage dl rows for this block (coalesced b128 loads)
  const float4* dlbase = (const float4*)(DL + ((size_t)b * 1024 + n0) * K_AGG);
#pragma unroll
  for (int i = 0; i < NPERB * (K_AGG / 4) / 256; ++i)
    sdl4[i * 256 + tid] = dlbase[i * 256 + tid];
  __syncthreads();

  // per-thread registers: this thread's cl row and the agg vector
  float cl[K_AGG], ag[K_AGG];
  const float4* clrow = (const float4*)(CL + ((size_t)b * 1024 + m) * K_AGG);
  const float4* agp   = (const float4*)aggv;
#pragma unroll
  for (int j = 0; j < K_AGG / 4; ++j) {
    float4 t = clrow[j];
    cl[4 * j + 0] = t.x; cl[4 * j + 1] = t.y;
    cl[4 * j + 2] = t.z; cl[4 * j + 3] = t.w;
    float4 u = agp[j];
    ag[4 * j + 0] = u.x; ag[4 * j + 1] = u.y;
    ag[4 * j + 2] = u.z; ag[4 * j + 3] = u.w;
  }

  float* orow = out + ((size_t)b * 1024 + n0) * 1024 + m;

  for (int n = 0; n < NPERB; ++n) {
    float a0 = 0.f, a1 = 0.f, a2 = 0.f, a3 = 0.f;
#pragma unroll
    for (int j = 0; j < K_AGG / 4; ++j) {
      float4 d = sdl4[n * (K_AGG / 4) + j];   // LDS broadcast read
      a0 += fast_tanh(cl[4 * j + 0] + d.x) * ag[4 * j + 0];
      a1 += fast_tanh(cl[4 * j + 1] + d.y) * ag[4 * j + 1];
      a2 += fast_tanh(cl[4 * j + 2] + d.z) * ag[4 * j + 2];
      a3 += fast_tanh(cl[4 * j + 3] + d.w) * ag[4 * j + 3];
    }
    orow[(size_t)n * 1024] = (a0 + a1) + (a2 + a3);
  }
}

// ---------------------------------------------------------------------------
// Phase 1: Y[row, a] = sum_k X[row, k] * W[a, k] + bias[a]
//   X: [ROWS, 512], W: [64, 512], Y: [ROWS, 64]
// One wave (32 threads) per 16x16 output tile, f32 WMMA, K-step 4.
//
// A layout (16x4 f32, MxK): lanes 0-15 -> M=0..15 ; VGPR0 = K=0 (lo half) /
// K=2 (hi half), VGPR1 = K=1 / K=3. So lane reads X[M=lane&15][kbase+2*hi+{0,1}].
// B layout (4x16 f32, KxN): lanes 0-15 -> K=0..1 rows, lanes 16-31 -> K=2..3,
// lane&15 = N. With B = W^T, lane reads W[a=lane&15][kbase+2*hi+{0,1}].
// C/D (16x16 f32): VGPR i -> row M = i + 8*(lane>>4), col N = lane&15.
// ---------------------------------------------------------------------------
__global__ __launch_bounds__(32) void linproj_wmma(
    const float* __restrict__ X, const float* __restrict__ W,
    const float* __restrict__ bias, float* __restrict__ Y) {
  const int lane   = threadIdx.x;        // 0..31 (wave32)
  const int lo     = lane & 15;
  const int hi     = lane >> 4;
  const int tile_m = blockIdx.x * 16;    // output row tile
  const int tile_a = blockIdx.y * 16;    // output col (feature) tile

  const float* arow = X + (size_t)(tile_m + lo) * K_LDIM + 2 * hi;
  const float* brow = W + (size_t)(tile_a + lo) * K_LDIM + 2 * hi;

  v8f c = {};
#pragma unroll 4
  for (int k = 0; k < K_LDIM; k += 4) {
    v2f a = *(const v2f*)(arow + k);   // 8-byte aligned
    v2f b = *(const v2f*)(brow + k);
    c = __builtin_amdgcn_wmma_f32_16x16x4_f32(
        /*neg_a=*/false, a, /*neg_b=*/false, b,
        /*c_mod=*/(short)0, c, /*reuse_a=*/false, /*reuse_b=*/false);
  }

  const float bv = bias[tile_a + lo];
#pragma unroll
  for (int i = 0; i < 8; ++i) {
    const int row = tile_m + i + 8 * hi;
    Y[(size_t)row * K_AGG + tile_a + lo] = c[i] + bv;
  }
}

// ---------------------------------------------------------------------------
extern "C" void kernel_launch(void* const* d_in, const int* in_sizes, int n_in,
                              void* d_out, int out_size, void* d_ws, size_t ws_size,
                              hipStream_t stream) {
  const float* data = (const float*)d_in[0];   // [4,1024,512]
  const float* crit = (const float*)d_in[1];   // [4,1024,512]
  const float* Wl   = (const float*)d_in[2];   // [64,512]
  const float* bl   = (const float*)d_in[3];   // [64]
  const float* Wr   = (const float*)d_in[4];   // [64,512]
  const float* br   = (const float*)d_in[5];   // [64]
  const float* agg  = (const float*)d_in[6];   // [64]
  float* out = (float*)d_out;                  // [4,1024,1024]

  float* dl = (float*)d_ws;                    // [4096,64]
  float* cl = dl + (size_t)ROWS * K_AGG;       // [4096,64]  (2 MB total)

  dim3 g1(ROWS / 16, K_AGG / 16);              // 256 x 4 tiles, 1 wave each
  linproj_wmma<<<g1, 32, 0, stream>>>(data, Wl, bl, dl);
  linproj_wmma<<<g1, 32, 0, stream>>>(crit, Wr, br, cl);

  dim3 g2(1024 / NPERB, 1024 / 256, 4);        // 16 x 4 x 4 = 256 blocks
  pairwise_tanh<<<g2, 256, 0, stream>>>(dl, cl, agg, out);
}